// Threshold_weights2_52699248721943
// MI455X (gfx1250) — compile-verified
//
#include <hip/hip_runtime.h>
#include <math.h>

typedef float v4f __attribute__((ext_vector_type(4)));

// Monotonic float <-> uint encoding so a u32 atomic max implements float max.
__device__ __forceinline__ unsigned fenc(float x) {
    unsigned u = __float_as_uint(x);
    return (u & 0x80000000u) ? ~u : (u | 0x80000000u);
}
__device__ __forceinline__ float fdec(unsigned e) {
    return (e & 0x80000000u) ? __uint_as_float(e ^ 0x80000000u)
                             : __uint_as_float(~e);
}

// Branchless top-2 insert: after this, m1 >= m2 and {m1,m2} are the two
// largest of the multiset seen so far (duplicate max -> m2 == m1).
__device__ __forceinline__ void upd(float v, float& m1, float& m2) {
    float hi = fmaxf(v, m1);
    m2 = fmaxf(m2, fminf(v, m1));
    m1 = hi;
}
// Merge two top-2 pairs (multiset-correct).
__device__ __forceinline__ void merge(float o1, float o2, float& m1, float& m2) {
    float hi = fmaxf(m1, o1);
    float lo = fminf(m1, o1);
    m1 = hi;
    m2 = fmaxf(lo, fmaxf(m2, o2));
}

__global__ void tw_init_gmax(unsigned* g) {
    *g = 0x007FFFFFu;   // fenc(-inf)
}

__global__ void tw_finalize(const unsigned* __restrict__ g, float* __restrict__ out) {
    out[0] = fdec(*g);
}

#define TW_NWAVES 8

__global__ void __launch_bounds__(32 * TW_NWAVES)
tw_margin_softmax(const float* __restrict__ o1p,
                  const float* __restrict__ o2p,
                  const float* __restrict__ mp,
                  const int*   __restrict__ targets,
                  float*       __restrict__ out,    // out[0]=scalar (finalize), out[1+3*row+k]
                  unsigned*    __restrict__ gmax,
                  int N, int C) {
    const int lane   = threadIdx.x & 31;
    const int wave   = threadIdx.x >> 5;
    const int nwaves = blockDim.x >> 5;
    const int row    = blockIdx.x * nwaves + wave;

    // Per-wave double-buffered staging: [wave][stage][matrix][lane] (16B each)
    __shared__ v4f sbuf[TW_NWAVES][2][3][32];
    __shared__ float smax[32];

    float rmax = -INFINITY;
    if (row < N) {
        const float* r1 = o1p + (size_t)row * C;
        const float* r2 = o2p + (size_t)row * C;
        const float* r3 = mp  + (size_t)row * C;
        const v4f* q1 = (const v4f*)r1;
        const v4f* q2 = (const v4f*)r2;
        const v4f* q3 = (const v4f*)r3;

        float a1 = -INFINITY, a2 = -INFINITY;   // outputs1 top-2
        float b1 = -INFINITY, b2 = -INFINITY;   // outputs2 top-2
        float c1 = -INFINITY, c2 = -INFINITY;   // mimic    top-2

        const int nv4     = C >> 2;              // float4 elements per row (250)
        const int nchunks = (nv4 + 31) >> 5;     // 32-lane chunks (8; last partial)

        const unsigned lbase       = (unsigned)(size_t)&sbuf[wave][0][0][lane];
        const unsigned matStride   = (unsigned)(sizeof(v4f) * 32);       // 512 B
        const unsigned stageStride = (unsigned)(sizeof(v4f) * 32 * 3);   // 1536 B

        // Issue one chunk (3 async b128 loads -> LDS, NT hint, no VGPR dest).
        auto issue = [&](int c, int stage) {
            const int i = (c << 5) + lane;       // float4 index within row
            if (i < nv4) {
                const unsigned l0 = lbase + (unsigned)stage * stageStride;
                asm volatile("global_load_async_to_lds_b128 %0, %1, off th:TH_LOAD_NT"
                             :: "v"(l0),                 "v"(q1 + i) : "memory");
                asm volatile("global_load_async_to_lds_b128 %0, %1, off th:TH_LOAD_NT"
                             :: "v"(l0 + matStride),     "v"(q2 + i) : "memory");
                asm volatile("global_load_async_to_lds_b128 %0, %1, off th:TH_LOAD_NT"
                             :: "v"(l0 + 2 * matStride), "v"(q3 + i) : "memory");
            }
        };

        if (nchunks > 0) issue(0, 0);
        for (int c = 0; c < nchunks; ++c) {
            const int stage = c & 1;
            if (c + 1 < nchunks) {
                issue(c + 1, stage ^ 1);
                // async loads complete in order: <=3 outstanding means chunk c
                // (its 3 ops + LDS writes) is fully done.
                asm volatile("s_wait_asynccnt 0x3" ::: "memory");
            } else {
                asm volatile("s_wait_asynccnt 0x0" ::: "memory");
            }
            const int i = (c << 5) + lane;
            if (i < nv4) {
                v4f v = sbuf[wave][stage][0][lane];
                v4f w = sbuf[wave][stage][1][lane];
                v4f u = sbuf[wave][stage][2][lane];
                upd(v.x, a1, a2); upd(v.y, a1, a2); upd(v.z, a1, a2); upd(v.w, a1, a2);
                upd(w.x, b1, b2); upd(w.y, b1, b2); upd(w.z, b1, b2); upd(w.w, b1, b2);
                upd(u.x, c1, c2); upd(u.y, c1, c2); upd(u.z, c1, c2); upd(u.w, c1, c2);
            }
        }
        for (int i = (nv4 << 2) + lane; i < C; i += 32) {   // tail (C % 4)
            upd(r1[i], a1, a2); upd(r2[i], b1, b2); upd(r3[i], c1, c2);
        }

        // wave32 butterfly reduction of the three top-2 pairs.
        #pragma unroll
        for (int off = 16; off; off >>= 1) {
            merge(__shfl_xor(a1, off, 32), __shfl_xor(a2, off, 32), a1, a2);
            merge(__shfl_xor(b1, off, 32), __shfl_xor(b2, off, 32), b1, b2);
            merge(__shfl_xor(c1, off, 32), __shfl_xor(c2, off, 32), c1, c2);
        }
        rmax = fmaxf(a1, b1);

        if (lane == 0) {
            const int tgt = targets[row];
            const float t1 = r1[tgt], t2 = r2[tgt], t3 = r3[tgt];
            // bit-exact: a1 is a pure fmax chain over loaded values
            const float d1 = (t1 == a1) ? (a1 - a2) : 0.0f;
            const float d2 = (t2 == b1) ? (b1 - b2) : 0.0f;
            const float dm = (t3 == c1) ? (c1 - c2) : 0.0f;
            const float p0 = d1 * 0.5f, p1 = d2 * 0.5f, p2 = dm * 0.5f; // /T, T=2
            const float mx = fmaxf(p0, fmaxf(p1, p2));
            const float e0 = expf(p0 - mx);
            const float e1 = expf(p1 - mx);
            const float e2 = expf(p2 - mx);
            const float inv = 1.0f / (e0 + e1 + e2);
            float* o = out + 1 + 3 * (size_t)row;
            o[0] = e0 * inv;
            o[1] = e1 * inv;
            o[2] = e2 * inv;
        }
    }

    // Block-level max of max(outputs1,outputs2) row maxima -> 1 atomic/block.
    if (lane == 0) smax[wave] = rmax;
    __syncthreads();
    if (wave == 0) {
        float v = (lane < nwaves) ? smax[lane] : -INFINITY;
        #pragma unroll
        for (int off = 16; off; off >>= 1)
            v = fmaxf(v, __shfl_xor(v, off, 32));
        if (lane == 0) atomicMax(gmax, fenc(v));
    }
}

extern "C" void kernel_launch(void* const* d_in, const int* in_sizes, int n_in,
                              void* d_out, int out_size, void* d_ws, size_t ws_size,
                              hipStream_t stream) {
    const float* o1 = (const float*)d_in[0];   // outputs1 [N,C]
    const float* o2 = (const float*)d_in[1];   // outputs2 [N,C]
    const float* mm = (const float*)d_in[2];   // mimic    [N,C]
    const int*   tg = (const int*)d_in[3];     // targets  [N]
    const int N = in_sizes[3];
    const int C = in_sizes[0] / N;

    float*    out  = (float*)d_out;            // [0]=max_preds, [1..]=softmax [N,3]
    unsigned* gmax = (unsigned*)d_ws;

    tw_init_gmax<<<1, 1, 0, stream>>>(gmax);

    dim3 block(32 * TW_NWAVES);                // 8 wave32 = 8 rows/block
    dim3 grid((N + TW_NWAVES - 1) / TW_NWAVES);
    tw_margin_softmax<<<grid, block, 0, stream>>>(o1, o2, mm, tg, out, gmax, N, C);

    tw_finalize<<<1, 1, 0, stream>>>(gmax, out);
}